// TextLevelGNN_24455543783858
// MI455X (gfx1250) — compile-verified
//
#include <hip/hip_runtime.h>
#include <hip/hip_bf16.h>
#include <math.h>

// ---------------------------------------------------------------------------
// Problem constants (from the reference)
// ---------------------------------------------------------------------------
#define BATCH 128
#define LEN   512
#define KNBR  6
#define DIM   300
#define DIM2  (DIM / 2)          // 150 float2 per embedding row
#define NCLS  20
#define LSPLIT 8                 // L is split 8 ways for occupancy
#define LCHUNK (LEN / LSPLIT)    // 64

typedef __attribute__((ext_vector_type(2))) float v2f;
typedef __attribute__((ext_vector_type(8))) float v8f;

// ---------------------------------------------------------------------------
// Phase 1: gather + gated weighted aggregation, partial-summed over an L chunk.
// Grid: (BATCH, LSPLIT); Block: 160 threads (5 wave32), lane t owns dims
// {2t, 2t+1} -> every embedding-row gather is a coalesced global_load_b64
// (float2), halving VMEM instruction count vs. one-dim-per-lane. node_emb rows
// are 1200 B (8 B aligned), so float2 access is legal.
// Per-l index/weight loads are wave-uniform broadcasts (L2-cached).
// partial layout: [LSPLIT][BATCH][DIM]
// ---------------------------------------------------------------------------
__global__ void __launch_bounds__(160)
gnn_aggregate_partial(const int* __restrict__ X,
                      const int* __restrict__ NX,
                      const float* __restrict__ node_emb,
                      const float* __restrict__ edge_w,
                      const float* __restrict__ node_w,
                      float* __restrict__ partial) {
    const int b = blockIdx.x;
    const int s = blockIdx.y;
    const int t = threadIdx.x;
    if (t >= DIM2) return;

    const float2* __restrict__ emb2 = (const float2*)node_emb;

    float accx = 0.0f, accy = 0.0f;
    const int lbase = s * LCHUNK;

    for (int li = 0; li < LCHUNK; ++li) {
        const int pos = b * LEN + lbase + li;
        const int x = X[pos];
        const int* nxp = NX + pos * KNBR;

        // neighbor aggregation: Mn = sum_k edge_w[NX_k] * node_emb[NX_k]
        float mnx = 0.0f, mny = 0.0f;
#pragma unroll
        for (int k = 0; k < KNBR; ++k) {
            const int nx = nxp[k];
            const float ew = edge_w[nx];             // (Nbig,1) -> flat index
            const float2 e = emb2[nx * DIM2 + t];    // global_load_b64
            mnx = fmaf(ew, e.x, mnx);
            mny = fmaf(ew, e.y, mny);
        }
        const float nn = node_w[x];
        const float2 r = emb2[x * DIM2 + t];
        // (1-nn)*Mn + nn*Rn
        accx += (1.0f - nn) * mnx + nn * r.x;
        accy += (1.0f - nn) * mny + nn * r.y;
    }
    float2* __restrict__ p2 = (float2*)partial;
    p2[(s * BATCH + b) * DIM2 + t] = make_float2(accx, accy);
}

// ---------------------------------------------------------------------------
// Phase 2: reduce the LSPLIT partials -> Xs (BATCH x DIM). Deterministic
// fixed-order summation (no float atomics), float2-vectorized.
// ---------------------------------------------------------------------------
__global__ void __launch_bounds__(256)
gnn_reduce_partials(const float* __restrict__ partial,
                    float* __restrict__ Xs) {
    const int i = blockIdx.x * blockDim.x + threadIdx.x;
    if (i >= BATCH * DIM2) return;
    const float2* __restrict__ p2 = (const float2*)partial;
    float sx = 0.0f, sy = 0.0f;
#pragma unroll
    for (int s = 0; s < LSPLIT; ++s) {
        const float2 v = p2[s * (BATCH * DIM2) + i];
        sx += v.x;
        sy += v.y;
    }
    ((float2*)Xs)[i] = make_float2(sx, sy);
}

// ---------------------------------------------------------------------------
// Phase 3: h = relu(Xs @ fc_W^T + b); out = softmax(h).
// Uses V_WMMA_F32_16X16X4_F32 (fp32 matrix core path, K-step 4, f32 accum),
// one wave per block, each block computes a 16-row x 32-col tile (cols 20..31
// are zero-weight padding), 75 K-steps over DIM=300.
//
// fp32 A-fragment layout (16x4): lanes 0-15 -> K={0,1}, lanes 16-31 -> K={2,3},
// row M = lane&15. B-fragment (4x16) mirrors it with col N = lane&15.
// C/D layout: acc[v] is row (v + 8*(lane>=16)), col (lane&15).
// ---------------------------------------------------------------------------
__global__ void __launch_bounds__(32)
gnn_fc_softmax_wmma(const float* __restrict__ Xs,
                    const float* __restrict__ fc_W,
                    const float* __restrict__ fc_b,
                    float* __restrict__ out) {
    const int lane = threadIdx.x;            // 0..31, one wave
    const int M0 = blockIdx.x * 16;          // 8 blocks cover 128 rows
    const int mrow = M0 + (lane & 15);
    const int kb = (lane >= 16) ? 2 : 0;     // K sub-offset within the 4-step
    const int ncol = lane & 15;
    const int n1 = 16 + ncol;                // second N tile (cols 16..31)
    const int n1c = (n1 < NCLS) ? n1 : (NCLS - 1);
    const float n1mask = (n1 < NCLS) ? 1.0f : 0.0f;

    v8f acc0 = {};
    v8f acc1 = {};

    for (int k0 = 0; k0 < DIM; k0 += 4) {
        v2f a;
        a.x = Xs[mrow * DIM + k0 + kb];
        a.y = Xs[mrow * DIM + k0 + kb + 1];

        v2f b0, b1;
        b0.x = fc_W[ncol * DIM + k0 + kb];
        b0.y = fc_W[ncol * DIM + k0 + kb + 1];
        b1.x = n1mask * fc_W[n1c * DIM + k0 + kb];
        b1.y = n1mask * fc_W[n1c * DIM + k0 + kb + 1];

        acc0 = __builtin_amdgcn_wmma_f32_16x16x4_f32(
            /*neg_a=*/false, a, /*neg_b=*/false, b0,
            /*c_mod=*/(short)0, acc0, /*reuse_a=*/false, /*reuse_b=*/false);
        acc1 = __builtin_amdgcn_wmma_f32_16x16x4_f32(
            /*neg_a=*/false, a, /*neg_b=*/false, b1,
            /*c_mod=*/(short)0, acc1, /*reuse_a=*/false, /*reuse_b=*/false);
    }

    // Spill the 16x32 tile to LDS with bias + relu applied.
    __shared__ float h[16][32];
    const int rbase = (lane >= 16) ? 8 : 0;
    const float bias0 = fc_b[ncol];                       // ncol < 16 < 20
    const float bias1 = (n1 < NCLS) ? fc_b[n1] : 0.0f;
#pragma unroll
    for (int v = 0; v < 8; ++v) {
        const int m = rbase + v;
        h[m][ncol]      = fmaxf(acc0[v] + bias0, 0.0f);
        h[m][16 + ncol] = fmaxf(acc1[v] + bias1, 0.0f);
    }
    __syncthreads();

    // Softmax over the 20 real classes, one lane per row.
    if (lane < 16) {
        const int m = lane;
        float mx = -1e30f;
#pragma unroll
        for (int c = 0; c < NCLS; ++c) mx = fmaxf(mx, h[m][c]);
        float e[NCLS];
        float ssum = 0.0f;
#pragma unroll
        for (int c = 0; c < NCLS; ++c) {
            e[c] = expf(h[m][c] - mx);
            ssum += e[c];
        }
        const float inv = 1.0f / ssum;
#pragma unroll
        for (int c = 0; c < NCLS; ++c)
            out[(M0 + m) * NCLS + c] = e[c] * inv;
    }
}

// ---------------------------------------------------------------------------
// Launch. d_in order: X, NX, EW(unused), node_emb, edge_w, node_w, fc_W, fc_b
// ---------------------------------------------------------------------------
extern "C" void kernel_launch(void* const* d_in, const int* in_sizes, int n_in,
                              void* d_out, int out_size, void* d_ws, size_t ws_size,
                              hipStream_t stream) {
    const int*   X        = (const int*)  d_in[0];
    const int*   NX       = (const int*)  d_in[1];
    // d_in[2] = EW, unused by the reference forward
    const float* node_emb = (const float*)d_in[3];
    const float* edge_w   = (const float*)d_in[4];
    const float* node_w   = (const float*)d_in[5];
    const float* fc_W     = (const float*)d_in[6];
    const float* fc_b     = (const float*)d_in[7];
    float*       out      = (float*)d_out;

    // Workspace: [LSPLIT][BATCH][DIM] partials, then [BATCH][DIM] Xs.
    float* partial = (float*)d_ws;
    float* Xs      = partial + (size_t)LSPLIT * BATCH * DIM;

    // Phase 1: gather + gated aggregation (L2-streaming float2 gathers)
    gnn_aggregate_partial<<<dim3(BATCH, LSPLIT), 160, 0, stream>>>(
        X, NX, node_emb, edge_w, node_w, partial);

    // Phase 2: deterministic partial reduction -> Xs
    gnn_reduce_partials<<<(BATCH * DIM2 + 255) / 256, 256, 0, stream>>>(
        partial, Xs);

    // Phase 3: FC via fp32 WMMA + softmax
    gnn_fc_softmax_wmma<<<BATCH / 16, 32, 0, stream>>>(
        Xs, fc_W, fc_b, out);
}